// PixelContrastLoss_8761733284019
// MI455X (gfx1250) — compile-verified
//
#include <hip/hip_runtime.h>
#include <math.h>
#include <stdint.h>

typedef __attribute__((ext_vector_type(2))) float v2f;
typedef __attribute__((ext_vector_type(8))) float v8f;
typedef __attribute__((ext_vector_type(4))) unsigned int u32x4;
typedef __attribute__((ext_vector_type(8))) int i32x8;
typedef __attribute__((ext_vector_type(4))) int i32x4;

#define HW        16384
#define CDIM      256
#define BIMG      8
#define NCLS      256
#define IGNORE_L  255
#define MAXSAMP   8192
#define MAXVIEW   256
#define MAXT      2048
#define MAXM      8192
#define TEMP      0.1f
#define BTEMP     0.07f
#define BROW      260          // LDS row stride in floats (256 + 4 pad -> bank-conflict-free)

// workspace layout (byte offsets)
#define O_META   0                                  // 16 ints: T,n_view,M,M16,hb,eb
#define O_ACC    64                                 // 1 float accumulator
#define O_CNT_N  128                                // BIMG*NCLS ints
#define O_CNT_H  (O_CNT_N + BIMG*NCLS*4)
#define O_ANCH   (O_CNT_H + BIMG*NCLS*4)            // MAXT ints  (img<<16|cls)
#define O_HK     (O_ANCH + MAXT*4)                  // MAXT ints
#define O_PIX    (O_HK + MAXT*4)                    // MAXM ints
#define O_CLS    (O_PIX + MAXM*4)                   // MAXM floats (row class, -1 = pad)
#define O_RM     (O_CLS + MAXM*4)                   // MAXM floats (row max)
#define O_RNS    (O_RM + MAXM*4)                    // MAXM floats (row neg-sum)
#define O_CF     (((O_RNS + MAXM*4) + 255) & ~255)  // MAXM*CDIM floats (gathered feats)

__device__ __forceinline__ v8f wmma_f32(v2f a, v2f b, v8f c) {
  // D = A(16x4,f32) * B(4x16,f32) + C(16x16,f32)  -> v_wmma_f32_16x16x4_f32
  return __builtin_amdgcn_wmma_f32_16x16x4_f32(false, a, false, b, (short)0, c,
                                               false, false);
}

// TDM: DMA one 16-row x 256-f32 tile (row-major, stride 256) from global into
// LDS with 16B padding per 1024B row (LDS row stride = 1040B = BROW floats).
// D# per CDNA5 ISA; groups 2/3 zero (2-D tensor).
__device__ __forceinline__ void tdm_load_16x256(const float* gsrc,
                                                unsigned lds_off) {
  unsigned long long ga = (unsigned long long)(uintptr_t)gsrc;
  u32x4 g0;
  g0[0] = 1u;                                       // count=1, user descriptor
  g0[1] = lds_off;                                  // lds_addr
  g0[2] = (unsigned)ga;                             // global_addr[31:0]
  g0[3] = (unsigned)((ga >> 32) & 0x01ffffffu)      // global_addr[56:32]
        | (2u << 30);                               // type=2 ("image")
  i32x8 g1;
  g1[0] = (2 << 16)        // data_size = 4B
        | (1 << 20)        // pad_enable
        | (7 << 22)        // pad_interval: 256 DWORDs (one 1024B row)
        | (3 << 25);       // pad_amount:   4 DWORDs (16B)
  g1[1] = (int)(256u << 16);   // tensor_dim0 = 256 (bits 79:48, low half)
  g1[2] = (int)(16u << 16);    // tensor_dim0 hi = 0 ; tensor_dim1 = 16
  g1[3] = (int)(256u << 16);   // tensor_dim1 hi = 0 ; tile_dim0 = 256
  g1[4] = 16;                  // tile_dim1 = 16 ; tile_dim2 = 0 (unused)
  g1[5] = 256;                 // tensor_dim0_stride = 256 (low 32)
  g1[6] = 0;                   // stride hi, tensor_dim1_stride lo
  g1[7] = 0;
  i32x4 z4 = {0, 0, 0, 0};
#if defined(__clang_major__) && __clang_major__ >= 23
  i32x8 z8 = {0, 0, 0, 0, 0, 0, 0, 0};
  __builtin_amdgcn_tensor_load_to_lds(g0, g1, z4, z4, z8, 0);
#else
  __builtin_amdgcn_tensor_load_to_lds(g0, g1, z4, z4, 0);
#endif
}

__global__ void k_zero(int* cnt_n, int* cnt_h, float* acc) {
  int i = blockIdx.x * blockDim.x + threadIdx.x;
  if (i < BIMG * NCLS) { cnt_n[i] = 0; cnt_h[i] = 0; }
  if (i == 0) *acc = 0.f;
}

__global__ void k_count(const int* __restrict__ yh, const int* __restrict__ yp,
                        int* cnt_n, int* cnt_h) {
  int g = blockIdx.x * blockDim.x + threadIdx.x;
  if (g >= BIMG * HW) return;
  int ii = g / HW;
  int c = yh[g];
  if ((unsigned)c < NCLS && c != IGNORE_L) {
    atomicAdd(&cnt_n[ii * NCLS + c], 1);
    if (yp[g] != c) atomicAdd(&cnt_h[ii * NCLS + c], 1);
  }
}

__global__ void k_plan(const int* cnt_n, const int* cnt_h, int* meta, int* anch,
                       int* hkarr) {
  int total = 0;
  for (int ii = 0; ii < BIMG; ++ii)
    for (int c = 0; c < NCLS; ++c)
      if (c != IGNORE_L && cnt_n[ii * NCLS + c] > MAXVIEW)
        anch[total++] = (ii << 16) | c;
  int n_view = 0, hb = 0, eb = 0;
  if (total > 0) {
    n_view = MAXSAMP / total; if (n_view > MAXVIEW) n_view = MAXVIEW;
    hb = n_view / 2; eb = n_view - hb;
    for (int t = 0; t < total; ++t) {
      int ii = anch[t] >> 16, c = anch[t] & 0xffff;
      int n  = cnt_n[ii * NCLS + c];
      int nh = cnt_h[ii * NCLS + c];
      int ne = n - nh;
      int hk;
      if (nh >= hb && ne >= eb)      hk = hb;
      else if (nh >= hb)             hk = n_view - ne;
      else if (ne >= eb)             hk = nh;
      else                           hk = (nh < n_view) ? nh : n_view;
      hkarr[t] = hk;
    }
  }
  int M = total * n_view;
  meta[0] = total; meta[1] = n_view; meta[2] = M;
  meta[3] = (M + 15) & ~15; meta[4] = hb; meta[5] = eb;
}

__global__ void k_select(const int* __restrict__ yh, const int* __restrict__ yp,
                         const int* meta, const int* anch, const int* hkarr,
                         int* pix) {
  int a = blockIdx.x;
  if (a >= meta[0]) return;
  if (threadIdx.x != 0) return;     // ordered first-k compaction: sequential scan
  int n_view = meta[1];
  int ii = anch[a] >> 16, c = anch[a] & 0xffff;
  int hk = hkarr[a], ek = n_view - hk;
  const int* Y = yh + ii * HW;
  const int* P = yp + ii * HW;
  int* out = pix + a * n_view;
  int hcnt = 0, ecnt = 0;
  for (int p = 0; p < HW && (hcnt < hk || ecnt < ek); ++p) {
    if (Y[p] != c) continue;
    if (P[p] != c) { if (hcnt < hk) out[hcnt++] = p; }
    else           { if (ecnt < ek) out[hk + ecnt++] = p; }
  }
}

__global__ void k_gather(const float* __restrict__ feats, const int* meta,
                         const int* anch, const int* pix, float* cf,
                         float* clsr) {
  int r = blockIdx.x;
  int M = meta[2], M16 = meta[3];
  if (r >= M16) return;
  int d = threadIdx.x;                         // 0..255 = feature channel
  if (r >= M) {                                // pad row: zeros, class -1
    cf[(size_t)r * CDIM + d] = 0.f;
    if (d == 0) clsr[r] = -1.f;
    return;
  }
  int T = meta[0], n_view = meta[1];
  int t = r % T, v = r / T;                    // cf row = v*T + t (ref layout)
  int ii = anch[t] >> 16, c = anch[t] & 0xffff;
  int p = pix[t * n_view + v];
  cf[(size_t)r * CDIM + d] = feats[((size_t)ii * CDIM + d) * HW + p];
  if (d == 0) clsr[r] = (float)c;
}

// Pass 1: per-row online max + rescaled negative-exp sum (flash-style).
// Block = 4 waves, each owning one 16-row tile; B tiles TDM-DMA'd into
// double-buffered LDS once per block and shared by all 4 waves.
__global__ __launch_bounds__(128) void k_pass1(const float* __restrict__ cf,
        const float* __restrict__ clsr, const int* meta,
        float* rowm, float* rowns) {
  __shared__ float bb[2][16 * BROW];
  int M = meta[2], M16 = meta[3];
  int tid = threadIdx.x;
  int w = tid >> 5, lane = tid & 31;
  int block_r0 = blockIdx.x * 64;
  if (block_r0 >= M16) return;                 // block-uniform exit (no barriers yet)
  int r0 = block_r0 + w * 16;
  int half = lane >> 4, ln = lane & 15;

  // Preload 16x256 f32 A-tile: lane holds K = 4k + 2*half + {0,1} of row r0+ln.
  // Rows >= M16 read in-bounds workspace garbage; results masked at the write.
  const float* arow = cf + (size_t)(r0 + ln) * CDIM + 2 * half;
  v2f a[64];
  #pragma unroll
  for (int k = 0; k < 64; ++k) a[k] = *(const v2f*)(arow + 4 * k);

  float rcls[8];
  #pragma unroll
  for (int d = 0; d < 8; ++d) rcls[d] = clsr[r0 + d + 8 * half];

  float m[8], s[8];
  #pragma unroll
  for (int d = 0; d < 8; ++d) { m[d] = -1e30f; s[d] = 0.f; }

  int ntile = M16 >> 4;
  unsigned lds0 = (unsigned)(uintptr_t)&bb[0][0];
  unsigned lds1 = (unsigned)(uintptr_t)&bb[1][0];
  if (w == 0) tdm_load_16x256(cf, lds0);       // prologue: tile 0 -> buf 0

  for (int jt = 0; jt < ntile; ++jt) {
    int cur = jt & 1;
    if (w == 0) {
      if (jt + 1 < ntile) {
        tdm_load_16x256(cf + (size_t)(jt + 1) * 16 * CDIM, cur ? lds0 : lds1);
        __builtin_amdgcn_s_wait_tensorcnt(1);  // tile jt landed in LDS
      } else {
        __builtin_amdgcn_s_wait_tensorcnt(0);
      }
    }
    __syncthreads();                           // publish buf[cur] to all waves

    int j = (jt << 4) + ln;
    float ccls  = clsr[j];
    bool cvalid = j < M;
    const float* bbase = &bb[cur][0] + ln * BROW + 2 * half;
    v8f acc = {0.f, 0.f, 0.f, 0.f, 0.f, 0.f, 0.f, 0.f};
    #pragma unroll
    for (int kc = 0; kc < 4; ++kc) {           // K=256 in 4 chunks of 64
      v2f b[16];
      #pragma unroll
      for (int i = 0; i < 16; ++i) b[i] = *(const v2f*)(bbase + 4 * (kc * 16 + i));
      #pragma unroll
      for (int i = 0; i < 16; ++i) acc = wmma_f32(a[kc * 16 + i], b[i], acc);
    }
    #pragma unroll
    for (int d = 0; d < 8; ++d) {              // C[d] = row (d+8*half), col ln
      float val = acc[d] * (1.f / TEMP);
      float mn  = fmaxf(m[d], cvalid ? val : -1e30f);
      float add = (cvalid && ccls != rcls[d]) ? __expf(val - mn) : 0.f;
      s[d] = s[d] * __expf(m[d] - mn) + add;
      m[d] = mn;
    }
    __syncthreads();                           // buf[cur] free for next DMA
  }
  // merge (m,s) across the 16 lanes holding each row's column subsets
  #pragma unroll
  for (int off = 1; off < 16; off <<= 1) {
    #pragma unroll
    for (int d = 0; d < 8; ++d) {
      float om = __shfl_xor(m[d], off);
      float os = __shfl_xor(s[d], off);
      float mn = fmaxf(m[d], om);
      s[d] = s[d] * __expf(m[d] - mn) + os * __expf(om - mn);
      m[d] = mn;
    }
  }
  if (ln == 0) {
    #pragma unroll
    for (int d = 0; d < 8; ++d) {
      int r = r0 + d + 8 * half;
      if (r < M) { rowm[r] = m[d]; rowns[r] = s[d]; }
    }
  }
}

// Pass 2: S = sum_pos (l - log(e^l + neg_sum)), P = pos count, accumulate loss.
__global__ __launch_bounds__(128) void k_pass2(const float* __restrict__ cf,
        const float* __restrict__ clsr, const int* meta,
        const float* __restrict__ rowm, const float* __restrict__ rowns,
        float* accum) {
  __shared__ float bb[2][16 * BROW];
  int M = meta[2], M16 = meta[3];
  int tid = threadIdx.x;
  int w = tid >> 5, lane = tid & 31;
  int block_r0 = blockIdx.x * 64;
  if (block_r0 >= M16) return;                 // block-uniform exit
  int r0 = block_r0 + w * 16;
  int half = lane >> 4, ln = lane & 15;

  const float* arow = cf + (size_t)(r0 + ln) * CDIM + 2 * half;
  v2f a[64];
  #pragma unroll
  for (int k = 0; k < 64; ++k) a[k] = *(const v2f*)(arow + 4 * k);

  float rcls[8], rm[8], rns[8], S[8], P[8];
  #pragma unroll
  for (int d = 0; d < 8; ++d) {
    int r = r0 + d + 8 * half;
    rcls[d] = clsr[r];
    rm[d]   = (r < M) ? rowm[r]  : 0.f;
    rns[d]  = (r < M) ? rowns[r] : 1.f;
    S[d] = 0.f; P[d] = 0.f;
  }

  int ntile = M16 >> 4;
  unsigned lds0 = (unsigned)(uintptr_t)&bb[0][0];
  unsigned lds1 = (unsigned)(uintptr_t)&bb[1][0];
  if (w == 0) tdm_load_16x256(cf, lds0);

  for (int jt = 0; jt < ntile; ++jt) {
    int cur = jt & 1;
    if (w == 0) {
      if (jt + 1 < ntile) {
        tdm_load_16x256(cf + (size_t)(jt + 1) * 16 * CDIM, cur ? lds0 : lds1);
        __builtin_amdgcn_s_wait_tensorcnt(1);
      } else {
        __builtin_amdgcn_s_wait_tensorcnt(0);
      }
    }
    __syncthreads();

    int j = (jt << 4) + ln;
    float ccls  = clsr[j];
    bool cvalid = j < M;
    const float* bbase = &bb[cur][0] + ln * BROW + 2 * half;
    v8f acc = {0.f, 0.f, 0.f, 0.f, 0.f, 0.f, 0.f, 0.f};
    #pragma unroll
    for (int kc = 0; kc < 4; ++kc) {
      v2f b[16];
      #pragma unroll
      for (int i = 0; i < 16; ++i) b[i] = *(const v2f*)(bbase + 4 * (kc * 16 + i));
      #pragma unroll
      for (int i = 0; i < 16; ++i) acc = wmma_f32(a[kc * 16 + i], b[i], acc);
    }
    #pragma unroll
    for (int d = 0; d < 8; ++d) {
      int r = r0 + d + 8 * half;
      float val = acc[d] * (1.f / TEMP);
      float l = val - rm[d];                   // <= 0 by construction
      bool pos = cvalid && (ccls == rcls[d]) && (j != r);
      float lp = l - __logf(__expf(l) + rns[d]);
      S[d] += pos ? lp : 0.f;
      P[d] += pos ? 1.f : 0.f;
    }
    __syncthreads();
  }
  #pragma unroll
  for (int off = 1; off < 16; off <<= 1) {
    #pragma unroll
    for (int d = 0; d < 8; ++d) {
      S[d] += __shfl_xor(S[d], off);
      P[d] += __shfl_xor(P[d], off);
    }
  }
  if (ln == 0) {
    float invM = 1.f / (float)M;
    #pragma unroll
    for (int d = 0; d < 8; ++d) {
      int r = r0 + d + 8 * half;
      if (r < M && P[d] > 0.f) {
        float lossr = -(TEMP / BTEMP) * (S[d] / P[d]);
        atomicAdd(accum, lossr * invM);
      }
    }
  }
}

__global__ void k_final(const int* meta, const float* accum, float* out) {
  if (blockIdx.x == 0 && threadIdx.x == 0)
    out[0] = (meta[0] == 0) ? 0.f : *accum;
}

extern "C" void kernel_launch(void* const* d_in, const int* in_sizes, int n_in,
                              void* d_out, int out_size, void* d_ws, size_t ws_size,
                              hipStream_t stream) {
  (void)in_sizes; (void)n_in; (void)out_size; (void)ws_size;
  const float* feats  = (const float*)d_in[0];
  const int*  labels  = (const int*)d_in[1];
  const int*  predict = (const int*)d_in[2];

  char* ws = (char*)d_ws;
  int*   meta  = (int*)(ws + O_META);
  float* acc   = (float*)(ws + O_ACC);
  int*   cnt_n = (int*)(ws + O_CNT_N);
  int*   cnt_h = (int*)(ws + O_CNT_H);
  int*   anch  = (int*)(ws + O_ANCH);
  int*   hkarr = (int*)(ws + O_HK);
  int*   pix   = (int*)(ws + O_PIX);
  float* clsr  = (float*)(ws + O_CLS);
  float* rowm  = (float*)(ws + O_RM);
  float* rowns = (float*)(ws + O_RNS);
  float* cf    = (float*)(ws + O_CF);

  k_zero  <<<(BIMG * NCLS + 255) / 256, 256, 0, stream>>>(cnt_n, cnt_h, acc);
  k_count <<<(BIMG * HW + 255) / 256, 256, 0, stream>>>(labels, predict, cnt_n, cnt_h);
  k_plan  <<<1, 1, 0, stream>>>(cnt_n, cnt_h, meta, anch, hkarr);
  k_select<<<MAXT, 32, 0, stream>>>(labels, predict, meta, anch, hkarr, pix);
  k_gather<<<MAXM, CDIM, 0, stream>>>(feats, meta, anch, pix, cf, clsr);
  // 512 row-tiles max, 4 waves (128 threads) per block -> 128 blocks
  k_pass1 <<<MAXM / 16 / 4, 128, 0, stream>>>(cf, clsr, meta, rowm, rowns);
  k_pass2 <<<MAXM / 16 / 4, 128, 0, stream>>>(cf, clsr, meta, rowm, rowns, acc);
  k_final <<<1, 1, 0, stream>>>(meta, acc, (float*)d_out);
}